// VQVAE_3899830305313
// MI455X (gfx1250) — compile-verified
//
#include <hip/hip_runtime.h>
#include <hip/hip_bf16.h>

// VQ-VAE forward, MI455X (gfx1250, wave32).
// All GEMMs use V_WMMA_F32_16X16X4_F32 (f32 in / f32 accumulate): the VQ
// argmin best/2nd-best gap (~5e-7) is far below bf16 input rounding error,
// so f32 matrix ops are required for index-exactness.
// B operands (embeddings + weights) are pre-packed in a K-paired layout so
// every WMMA B-fragment is a single coalesced global_load_b64.

#define NROWS 131072
#define DDIM  256
#define KEMB  2048
#define FDIM  512

typedef __attribute__((ext_vector_type(2))) float v2f;
typedef __attribute__((ext_vector_type(8))) float v8f;

__device__ __forceinline__ v8f wmma_f32x4(v2f a, v2f b, v8f c) {
  // 8 args: (neg_a, A, neg_b, B, c_mod, C, reuse_a, reuse_b)
  return __builtin_amdgcn_wmma_f32_16x16x4_f32(false, a, false, b, (short)0, c,
                                               false, false);
}

// ---------------------------------------------------------------------------
// Packed-B layout: P[(k>>1) * (2*N) + n*2 + (k&1)] = B[k*N + n].
// For even koff the (koff, koff+1) pair for column n sits at
// P + koff*N + n*2 .. +1  -> one 8-byte-aligned b64 load per fragment.
// ---------------------------------------------------------------------------
__global__ void pack_w(const float* __restrict__ W, float* __restrict__ Wp,
                       int K, int NOUT) {
  const int i = blockIdx.x * 256 + threadIdx.x;
  if (i < K * NOUT) {
    const int k = i / NOUT, n = i % NOUT;
    Wp[(size_t)(k >> 1) * (2 * NOUT) + n * 2 + (k & 1)] = W[i];
  }
}

// ---------------------------------------------------------------------------
// Per-wave 16x16 tile accumulation: C_tile += A(MxK) * Wp(K-paired, N cols).
// A fragment: lane covers row M=lane%16, Ks koff=k0+2*(lane/16)+{0,1} (b64).
// B fragment: lane covers col N=lane%16, same K split (single b64, coalesced).
// ---------------------------------------------------------------------------
template <int K, int NOUT>
__device__ __forceinline__ v8f gemm_tile_acc(const float* __restrict__ A,
                                             const float* __restrict__ Wp,
                                             int tM, int tN, int lrow, int lhalf) {
  v8f acc = {0.f, 0.f, 0.f, 0.f, 0.f, 0.f, 0.f, 0.f};
  const float* __restrict__ arow = A + (size_t)(tM * 16 + lrow) * K;
  const float* __restrict__ wcol = Wp + (size_t)(tN * 16 + lrow) * 2;
#pragma unroll 8
  for (int k0 = 0; k0 < K; k0 += 4) {
    const int koff = k0 + 2 * lhalf;  // always even
    const v2f a = *(const v2f*)(arow + koff);
    const v2f b = *(const v2f*)(wcol + (size_t)koff * NOUT);
    acc = wmma_f32x4(a, b, acc);
  }
  return acc;
}

// out = relu(A @ W + bias); block = 8 waves = 8 tiles.
template <int K, int NOUT>
__global__ void gemm_bias_relu(const float* __restrict__ A,
                               const float* __restrict__ Wp,
                               const float* __restrict__ bias,
                               float* __restrict__ out) {
  const int lane = threadIdx.x & 31;
  const int wv = threadIdx.x >> 5;
  const int tile = blockIdx.x * 8 + wv;
  const int tilesN = NOUT / 16;
  const int tN = tile % tilesN;
  const int tM = tile / tilesN;
  const int lrow = lane & 15, lhalf = lane >> 4;

  v8f acc = gemm_tile_acc<K, NOUT>(A, Wp, tM, tN, lrow, lhalf);

  const int n = tN * 16 + lrow;
  const float bv = bias[n];
#pragma unroll
  for (int j = 0; j < 8; ++j) {
    const int m = tM * 16 + j + 8 * lhalf;
    float v = acc[j] + bv;
    v = fmaxf(v, 0.0f);
    out[(size_t)m * NOUT + n] = v;
  }
}

// Final decoder layer fused with recon loss: accumulates sum((relu(..)-ref)^2),
// never materializes x_recon.
template <int K, int NOUT>
__global__ void gemm_bias_relu_l2(const float* __restrict__ A,
                                  const float* __restrict__ Wp,
                                  const float* __restrict__ bias,
                                  const float* __restrict__ ref,
                                  double* __restrict__ racc) {
  const int lane = threadIdx.x & 31;
  const int wv = threadIdx.x >> 5;
  const int tile = blockIdx.x * 8 + wv;
  const int tilesN = NOUT / 16;
  const int tN = tile % tilesN;
  const int tM = tile / tilesN;
  const int lrow = lane & 15, lhalf = lane >> 4;

  v8f acc = gemm_tile_acc<K, NOUT>(A, Wp, tM, tN, lrow, lhalf);

  const int n = tN * 16 + lrow;
  const float bv = bias[n];
  float rsum = 0.0f;
#pragma unroll
  for (int j = 0; j < 8; ++j) {
    const int m = tM * 16 + j + 8 * lhalf;
    const float v = fmaxf(acc[j] + bv, 0.0f);
    const float d = v - ref[(size_t)m * NOUT + n];
    rsum = fmaf(d, d, rsum);
  }
#pragma unroll
  for (int off = 16; off >= 1; off >>= 1) rsum += __shfl_xor(rsum, off, 32);
  if (lane == 0) atomicAdd(racc, (double)rsum);
}

// ---------------------------------------------------------------------------
// prep: embP (K-paired transpose of emb), e_sq[k] = ||emb_k||^2, zero accums.
// One block (256 thr) per embedding row; t indexes the latent dim d.
// embP[(d>>1)*2*KEMB + k*2 + (d&1)] = emb[k*DDIM + d].
// ---------------------------------------------------------------------------
__global__ void prep_kernel(const float* __restrict__ emb,
                            float* __restrict__ embP,
                            float* __restrict__ e_sq,
                            double* __restrict__ accums) {
  const int k = blockIdx.x;
  const int t = threadIdx.x;
  const float v = emb[(size_t)k * DDIM + t];
  embP[(size_t)(t >> 1) * (2 * KEMB) + k * 2 + (t & 1)] = v;
  float s = v * v;
#pragma unroll
  for (int off = 16; off >= 1; off >>= 1) s += __shfl_xor(s, off, 32);
  __shared__ float ws[8];
  if ((t & 31) == 0) ws[t >> 5] = s;
  __syncthreads();
  if (t == 0) {
    float tot = 0.f;
#pragma unroll
    for (int i = 0; i < 8; ++i) tot += ws[i];
    e_sq[k] = tot;
    if (k == 0) { accums[0] = 0.0; accums[1] = 0.0; }
  }
}

// ---------------------------------------------------------------------------
// dist_gather: per 16-row strip of z_e, find argmin_k(e_sq[k] - 2 z.e_k)
// (z_sq is row-constant -> drops out of argmin), gather z_q, emit
// z_st = z_e + (z_q - z_e) (matches reference STE arithmetic exactly),
// accumulate sum((z_q - z_e)^2).
// Block = 256 thr = 8 waves; wave w scans embedding cols [w*256, w*256+256).
// ---------------------------------------------------------------------------
__global__ void dist_gather(const float* __restrict__ z_e,
                            const float* __restrict__ embP,
                            const float* __restrict__ e_sq,
                            const float* __restrict__ emb,
                            float* __restrict__ z_out,
                            double* __restrict__ qacc) {
  __shared__ float sA[16 * 260];  // +4 pad kills stride-256 bank conflicts
  __shared__ float wbest[8][16];
  __shared__ int windex[8][16];
  __shared__ int fidx[16];

  const int t = threadIdx.x;
  const int rowbase = blockIdx.x * 16;

  // Stage 16x256 z_e strip into LDS (coalesced).
#pragma unroll
  for (int i = 0; i < 16; ++i) {
    const int idx = t + i * 256;
    const int r = idx >> 8, c = idx & 255;
    sA[r * 260 + c] = z_e[(size_t)(rowbase + r) * DDIM + c];
  }
  __syncthreads();

  const int w = t >> 5, lane = t & 31;
  const int lrow = lane & 15, lhalf = lane >> 4;

  float best[8];
  int bidx[8];
#pragma unroll
  for (int j = 0; j < 8; ++j) { best[j] = 3.4e38f; bidx[j] = 0; }

  const float* __restrict__ sArow = sA + lrow * 260;  // 8B aligned (260 even)

  for (int ct = 0; ct < 16; ++ct) {
    const int col = w * 256 + ct * 16 + lrow;
    const float* __restrict__ bcol = embP + (size_t)col * 2;
    v8f acc = {0.f, 0.f, 0.f, 0.f, 0.f, 0.f, 0.f, 0.f};
#pragma unroll 8
    for (int k0 = 0; k0 < DDIM; k0 += 4) {
      const int koff = k0 + 2 * lhalf;  // even
      const v2f a = *(const v2f*)(sArow + koff);
      const v2f b = *(const v2f*)(bcol + (size_t)koff * KEMB);
      acc = wmma_f32x4(a, b, acc);
    }
    const float es = e_sq[col];
#pragma unroll
    for (int j = 0; j < 8; ++j) {
      const float dv = es - 2.0f * acc[j];
      if (dv < best[j]) { best[j] = dv; bidx[j] = col; }  // strict <: first-min
    }
  }

  // Reduce across the 16 column-lanes of each half-wave (lowest idx on ties).
#pragma unroll
  for (int j = 0; j < 8; ++j) {
    float v = best[j];
    int idx = bidx[j];
#pragma unroll
    for (int off = 8; off >= 1; off >>= 1) {
      const float v2 = __shfl_xor(v, off, 16);
      const int i2 = __shfl_xor(idx, off, 16);
      if (v2 < v || (v2 == v && i2 < idx)) { v = v2; idx = i2; }
    }
    if (lrow == 0) {
      const int r = j + 8 * lhalf;
      wbest[w][r] = v;
      windex[w][r] = idx;
    }
  }
  __syncthreads();

  if (t < 16) {
    float v = wbest[0][t];
    int idx = windex[0][t];
#pragma unroll
    for (int ww = 1; ww < 8; ++ww) {
      const float v2 = wbest[ww][t];
      const int i2 = windex[ww][t];
      if (v2 < v || (v2 == v && i2 < idx)) { v = v2; idx = i2; }
    }
    fidx[t] = idx;
  }
  __syncthreads();

  // Gather + straight-through output + quantization-loss partial. t = dim.
  float qsum = 0.0f;
#pragma unroll 4
  for (int r = 0; r < 16; ++r) {
    const int e = fidx[r];
    const float q = emb[(size_t)e * DDIM + t];
    const float ze = sA[r * 260 + t];
    const float dqe = q - ze;
    z_out[(size_t)(rowbase + r) * DDIM + t] = ze + dqe;  // == z_q_st forward
    qsum = fmaf(dqe, dqe, qsum);
  }
#pragma unroll
  for (int off = 16; off >= 1; off >>= 1) qsum += __shfl_xor(qsum, off, 32);
  if (lane == 0) atomicAdd(qacc, (double)qsum);
}

__global__ void finalize_kernel(const double* __restrict__ accums,
                                float* __restrict__ loss_out) {
  const double quant = 1.25 * accums[0] / ((double)NROWS * (double)DDIM);
  const double recon = accums[1] / ((double)NROWS * (double)FDIM);
  loss_out[0] = (float)(recon + quant);
}

// ---------------------------------------------------------------------------
extern "C" void kernel_launch(void* const* d_in, const int* in_sizes, int n_in,
                              void* d_out, int out_size, void* d_ws, size_t ws_size,
                              hipStream_t stream) {
  (void)in_sizes; (void)n_in; (void)out_size; (void)ws_size;
  const float* x      = (const float*)d_in[0];
  const float* enc_w1 = (const float*)d_in[1];
  const float* enc_b1 = (const float*)d_in[2];
  const float* enc_w2 = (const float*)d_in[3];
  const float* enc_b2 = (const float*)d_in[4];
  const float* enc_w3 = (const float*)d_in[5];
  const float* enc_b3 = (const float*)d_in[6];
  const float* dec_w1 = (const float*)d_in[7];
  const float* dec_b1 = (const float*)d_in[8];
  const float* dec_w2 = (const float*)d_in[9];
  const float* dec_b2 = (const float*)d_in[10];
  const float* dec_w3 = (const float*)d_in[11];
  const float* dec_b3 = (const float*)d_in[12];
  const float* emb    = (const float*)d_in[13];

  float* out = (float*)d_out;               // [N*D z_latent][1 loss]
  float* loss_out = out + (size_t)NROWS * DDIM;

  char* ws = (char*)d_ws;
  float*  embP   = (float*)(ws);                       // 2 MB (K-paired embT)
  float*  e_sq   = (float*)(ws + (2u << 20));          // 8 KB
  double* accums = (double*)(ws + (2u << 20) + 8192);  // [0]=quant, [1]=recon
  // Packed weights (K-paired), 576 KB total, starting at 2MB+16KB.
  float* wp = (float*)(ws + (2u << 20) + 16384);
  float* p_ew1 = wp;                  // 512*64
  float* p_ew2 = p_ew1 + 512 * 64;    // 64*128
  float* p_ew3 = p_ew2 + 64 * 128;    // 128*256
  float* p_dw1 = p_ew3 + 128 * 256;   // 256*128
  float* p_dw2 = p_dw1 + 256 * 128;   // 128*64
  float* p_dw3 = p_dw2 + 128 * 64;    // 64*512
  // Region A (134 MB): h1 -> z_e -> d2.  Region B (67 MB): h2 -> d1.
  char* regA = ws + (4u << 20);
  char* regB = regA + ((size_t)NROWS * DDIM * sizeof(float));
  float* h1  = (float*)regA;  // N x 64
  float* h2  = (float*)regB;  // N x 128
  float* z_e = (float*)regA;  // N x 256 (h1 dead)
  float* d1  = (float*)regB;  // N x 128 (h2 dead)
  float* d2  = (float*)regA;  // N x 64  (z_e dead)

  const int TM = NROWS / 16;  // 8192 row tiles

  prep_kernel<<<KEMB, 256, 0, stream>>>(emb, embP, e_sq, accums);
  pack_w<<<(512 * 64 + 255) / 256, 256, 0, stream>>>(enc_w1, p_ew1, 512, 64);
  pack_w<<<(64 * 128 + 255) / 256, 256, 0, stream>>>(enc_w2, p_ew2, 64, 128);
  pack_w<<<(128 * 256 + 255) / 256, 256, 0, stream>>>(enc_w3, p_ew3, 128, 256);
  pack_w<<<(256 * 128 + 255) / 256, 256, 0, stream>>>(dec_w1, p_dw1, 256, 128);
  pack_w<<<(128 * 64 + 255) / 256, 256, 0, stream>>>(dec_w2, p_dw2, 128, 64);
  pack_w<<<(64 * 512 + 255) / 256, 256, 0, stream>>>(dec_w3, p_dw3, 64, 512);

  gemm_bias_relu<512, 64><<<TM * (64 / 16) / 8, 256, 0, stream>>>(x, p_ew1, enc_b1, h1);
  gemm_bias_relu<64, 128><<<TM * (128 / 16) / 8, 256, 0, stream>>>(h1, p_ew2, enc_b2, h2);
  gemm_bias_relu<128, 256><<<TM * (256 / 16) / 8, 256, 0, stream>>>(h2, p_ew3, enc_b3, z_e);

  dist_gather<<<TM, 256, 0, stream>>>(z_e, embP, e_sq, emb, out, accums + 0);

  gemm_bias_relu<256, 128><<<TM * (128 / 16) / 8, 256, 0, stream>>>(out, p_dw1, dec_b1, d1);
  gemm_bias_relu<128, 64><<<TM * (64 / 16) / 8, 256, 0, stream>>>(d1, p_dw2, dec_b2, d2);
  gemm_bias_relu_l2<64, 512><<<TM * (512 / 16) / 8, 256, 0, stream>>>(d2, p_dw3, dec_b3, x, accums + 1);

  finalize_kernel<<<1, 1, 0, stream>>>(accums, loss_out);
}